// GHMCLoss_69063074119704
// MI455X (gfx1250) — compile-verified
//
#include <hip/hip_runtime.h>

#define BINS 10
#define BLOCK 256
#define PAIRS_PER_THREAD 8   // 8 float4 iterations/thread -> 32 bin increments, < 63 (6-bit counters)
#define MAX_BLOCKS 8192

typedef __attribute__((ext_vector_type(2))) float v2f;
typedef __attribute__((ext_vector_type(8))) float v8f;

// Exact replica of searchsorted(edges, g, side='right') - 1 for edges = {0, .1f, ..., .9f, 1+1e-6f}
// All g here lie in [0, 1], so bin = number of interior edges <= g; always "valid".
__device__ __forceinline__ int bin_of(float g) {
    int b = 0;
    b += (g >= 0.1f); b += (g >= 0.2f); b += (g >= 0.3f);
    b += (g >= 0.4f); b += (g >= 0.5f); b += (g >= 0.6f);
    b += (g >= 0.7f); b += (g >= 0.8f); b += (g >= 0.9f);
    return b;  // 0..9
}

// Deterministic 32-lane (wave32) sum. Uses V_WMMA_F32_16X16X4_F32 with B = ones:
// A[m][k]: lanes 0-15 hold K=0 (=x), lanes 16-31 hold K=2 (=x); K=1,3 are 0.
// D[m][n] = x_m + x_{m+16} for every n; summing the 8 D VGPRs per lane gives the
// half-wave sums; one xor-16 shuffle completes the 32-lane total (same in all lanes).
__device__ __forceinline__ float wave_reduce_sum(float x) {
#if __has_builtin(__builtin_amdgcn_wmma_f32_16x16x4_f32)
    v2f a; a.x = x;    a.y = 0.0f;
    v2f b; b.x = 1.0f; b.y = 1.0f;
    v8f c = {};
    c = __builtin_amdgcn_wmma_f32_16x16x4_f32(false, a, false, b, (short)0, c, false, false);
    float s = ((c[0] + c[1]) + (c[2] + c[3])) + ((c[4] + c[5]) + (c[6] + c[7]));
    s += __shfl_xor(s, 16);
    return s;
#else
    for (int off = 16; off > 0; off >>= 1) x += __shfl_xor(x, off);
    return x;
#endif
}

__device__ __forceinline__ void flush_packed_hist(unsigned long long acc, unsigned* lds_hist) {
    #pragma unroll
    for (int b = 0; b < BINS; ++b) {
        unsigned c = (unsigned)((acc >> (6 * b)) & 63ull);
        if (c) atomicAdd(&lds_hist[b], c);
    }
}

__global__ void ghm_init_kernel(unsigned* __restrict__ counts) {
    if (threadIdx.x < BINS) counts[threadIdx.x] = 0u;
}

// Pass 1: histogram of g over all 2N elements. 16B loads for pred (2 items) and target (2 x int64).
__global__ __launch_bounds__(BLOCK) void ghm_hist_kernel(
    const float* __restrict__ pred, const long long* __restrict__ target,
    unsigned* __restrict__ counts, long long npairs, long long nitems)
{
    __shared__ unsigned lds_hist[BINS];
    if (threadIdx.x < BINS) lds_hist[threadIdx.x] = 0u;
    __syncthreads();

    const float4* __restrict__ pred4 = (const float4*)pred;
    const int4*   __restrict__ tgt2  = (const int4*)target;  // .x/.z = low words of two int64s

    const long long total = (long long)gridDim.x * BLOCK;
    const long long tid   = (long long)blockIdx.x * BLOCK + threadIdx.x;

    unsigned long long acc = 0ull;   // 10 bins x 6 bits
    int burst = 0;
    for (long long p = tid; p < npairs; p += total) {
        float4 pv = pred4[p];
        int4   tv = tgt2[p];
        float oh0 = (tv.x == 0) ? 1.0f : 0.0f;
        float oh1 = (tv.z == 0) ? 1.0f : 0.0f;
        acc += 1ull << (6 * bin_of(fabsf(pv.x - oh0)));
        acc += 1ull << (6 * bin_of(fabsf(pv.y - (1.0f - oh0))));
        acc += 1ull << (6 * bin_of(fabsf(pv.z - oh1)));
        acc += 1ull << (6 * bin_of(fabsf(pv.w - (1.0f - oh1))));
        if (++burst >= 14) { flush_packed_hist(acc, lds_hist); acc = 0ull; burst = 0; }
    }
    if (tid == 0 && (nitems & 1)) {  // odd tail item (not hit for the reference shape)
        long long i = nitems - 1;
        float p0 = pred[2 * i], p1 = pred[2 * i + 1];
        int t = (int)target[i];
        float oh = (t == 0) ? 1.0f : 0.0f;
        acc += 1ull << (6 * bin_of(fabsf(p0 - oh)));
        acc += 1ull << (6 * bin_of(fabsf(p1 - (1.0f - oh))));
    }
    flush_packed_hist(acc, lds_hist);
    __syncthreads();
    if (threadIdx.x < BINS) {
        unsigned c = lds_hist[threadIdx.x];
        if (c) atomicAdd(&counts[threadIdx.x], c);  // integer add: order-independent => deterministic
    }
}

// Pass 2: weighted loss; per-block partial sums (deterministic tree, no float atomics).
__global__ __launch_bounds__(BLOCK) void ghm_loss_kernel(
    const float* __restrict__ pred, const long long* __restrict__ target,
    const unsigned* __restrict__ counts, float* __restrict__ partials,
    long long npairs, long long nitems)
{
    __shared__ float wb[BINS];
    __shared__ float wsum8[BLOCK / 32];
    if (threadIdx.x == 0) {
        int n_ne = 0;
        unsigned c[BINS];
        #pragma unroll
        for (int b = 0; b < BINS; ++b) { c[b] = counts[b]; n_ne += (c[b] > 0u) ? 1 : 0; }
        float inv_nne = (n_ne > 0) ? (1.0f / (float)n_ne) : 1.0f;
        #pragma unroll
        for (int b = 0; b < BINS; ++b)
            wb[b] = (c[b] > 0u) ? (0.5f * inv_nne / (float)c[b]) : 0.0f;
        // per-item contribution = -pred[i,t] * (wb[b0] + wb[b1]); the /N and tot=N cancel.
    }
    __syncthreads();

    const float4* __restrict__ pred4 = (const float4*)pred;
    const int4*   __restrict__ tgt2  = (const int4*)target;

    const long long total = (long long)gridDim.x * BLOCK;
    const long long tid   = (long long)blockIdx.x * BLOCK + threadIdx.x;

    float acc = 0.0f;
    for (long long p = tid; p < npairs; p += total) {
        float4 pv = pred4[p];
        int4   tv = tgt2[p];
        float oh0 = (tv.x == 0) ? 1.0f : 0.0f;
        float oh1 = (tv.z == 0) ? 1.0f : 0.0f;
        int b00 = bin_of(fabsf(pv.x - oh0));
        int b01 = bin_of(fabsf(pv.y - (1.0f - oh0)));
        int b10 = bin_of(fabsf(pv.z - oh1));
        int b11 = bin_of(fabsf(pv.w - (1.0f - oh1)));
        float loss0 = -((tv.x == 0) ? pv.x : pv.y);
        float loss1 = -((tv.z == 0) ? pv.z : pv.w);
        acc += loss0 * (wb[b00] + wb[b01]);
        acc += loss1 * (wb[b10] + wb[b11]);
    }
    if (tid == 0 && (nitems & 1)) {
        long long i = nitems - 1;
        float p0 = pred[2 * i], p1 = pred[2 * i + 1];
        int t = (int)target[i];
        float oh = (t == 0) ? 1.0f : 0.0f;
        int b0 = bin_of(fabsf(p0 - oh));
        int b1 = bin_of(fabsf(p1 - (1.0f - oh)));
        acc += -((t == 0) ? p0 : p1) * (wb[b0] + wb[b1]);
    }

    float w = wave_reduce_sum(acc);                 // v_wmma_f32_16x16x4_f32 path
    if ((threadIdx.x & 31) == 0) wsum8[threadIdx.x >> 5] = w;
    __syncthreads();
    if (threadIdx.x == 0) {
        float s = 0.0f;
        #pragma unroll
        for (int i = 0; i < BLOCK / 32; ++i) s += wsum8[i];
        partials[blockIdx.x] = s;                   // fixed order => deterministic
    }
}

__global__ __launch_bounds__(BLOCK) void ghm_final_kernel(
    const float* __restrict__ partials, int n, float* __restrict__ out)
{
    __shared__ float wsum8[BLOCK / 32];
    float acc = 0.0f;
    for (int i = threadIdx.x; i < n; i += BLOCK) acc += partials[i];
    float w = wave_reduce_sum(acc);
    if ((threadIdx.x & 31) == 0) wsum8[threadIdx.x >> 5] = w;
    __syncthreads();
    if (threadIdx.x == 0) {
        float s = 0.0f;
        #pragma unroll
        for (int i = 0; i < BLOCK / 32; ++i) s += wsum8[i];
        out[0] = s;
    }
}

extern "C" void kernel_launch(void* const* d_in, const int* in_sizes, int n_in,
                              void* d_out, int out_size, void* d_ws, size_t ws_size,
                              hipStream_t stream) {
    const float*     pred   = (const float*)d_in[0];
    const long long* target = (const long long*)d_in[1];  // jnp int64
    long long nitems = (long long)in_sizes[1];            // N
    long long npairs = nitems >> 1;

    unsigned* counts   = (unsigned*)d_ws;                 // 10 x u32 at ws[0]
    float*    partials = (float*)((char*)d_ws + 64);      // per-block partial sums
    float*    out      = (float*)d_out;

    long long want = (npairs + (long long)BLOCK * PAIRS_PER_THREAD - 1) /
                     ((long long)BLOCK * PAIRS_PER_THREAD);
    if (want < 1) want = 1;
    int blocks = (int)((want < (long long)MAX_BLOCKS) ? want : (long long)MAX_BLOCKS);
    // (For the reference shape: npairs = 8388608 -> 4096 blocks, 8 pairs/thread.)

    ghm_init_kernel<<<1, 32, 0, stream>>>(counts);
    ghm_hist_kernel<<<blocks, BLOCK, 0, stream>>>(pred, target, counts, npairs, nitems);
    ghm_loss_kernel<<<blocks, BLOCK, 0, stream>>>(pred, target, counts, partials, npairs, nitems);
    ghm_final_kernel<<<1, BLOCK, 0, stream>>>(partials, blocks, out);
}